// PyramidDownSampling_84241488543967
// MI455X (gfx1250) — compile-verified
//
#include <hip/hip_runtime.h>
#include <stdint.h>

// Fused 3D Gaussian pyramid for MI455X (gfx1250).
// out = concat(x_copy, blur3_stride2(x), blur5_stride4(x)), fp32.
// One workgroup stages a 20^3 edge-clamped tile of x into LDS via the CDNA5
// async global->LDS data mover (ASYNCcnt), then produces all three outputs
// from that single staging pass (minimum-traffic fusion, ~287 MB total
// -> ~12.3 us at 23.3 TB/s; compute is ~0.36 GFLOP fp32, negligible).

#define TILE 16
#define HALO 2
#define TLD  (TILE + 2 * HALO)   // 20

#if __has_builtin(__builtin_amdgcn_global_load_async_to_lds_b32)
#define USE_ASYNC 1
#else
#define USE_ASYNC 0
#endif

typedef __attribute__((address_space(1))) int* as1_i32;
typedef __attribute__((address_space(3))) int* as3_i32;

__device__ __forceinline__ int clampi(int v, int lo, int hi) {
  return v < lo ? lo : (v > hi ? hi : v);
}

__global__ __launch_bounds__(256)
void pyr_fused_kernel(const float* __restrict__ x,
                      const float* __restrict__ k0,   // 3x3x3 (symmetric)
                      const float* __restrict__ k1,   // 5x5x5 (symmetric)
                      float* __restrict__ out_x,
                      float* __restrict__ out_d1,
                      float* __restrict__ out_d2)
{
  __shared__ float tile[TLD * TLD * TLD];  // 8000 floats = 32 KB
  __shared__ float k0s[27];
  __shared__ float k1s[125];

  const int tid   = threadIdx.x;
  const int bw    = blockIdx.x;        // W tile
  const int bh    = blockIdx.y;        // H tile
  const int bz    = blockIdx.z & 15;   // D tile
  const int batch = blockIdx.z >> 4;   // 0..1

  const int ow = bw * TILE, oh = bh * TILE, od = bz * TILE;
  const float* xb = x + (size_t)batch * (256ull * 256ull * 256ull);

  if (tid < 27) k0s[tid] = k0[tid];
  if (tid >= 32 && tid < 32 + 125) k1s[tid - 32] = k1[tid - 32];

  // ---- Stage 20^3 halo tile (edge-clamped == replicate pad) into LDS ----
  for (int i = tid; i < TLD * TLD * TLD; i += 256) {
    int lx = i % TLD;
    int t  = i / TLD;
    int ly = t % TLD;
    int lz = t / TLD;
    int gx = clampi(ow + lx - HALO, 0, 255);
    int gy = clampi(oh + ly - HALO, 0, 255);
    int gz = clampi(od + lz - HALO, 0, 255);
    const float* src = xb + (((size_t)gz * 256 + (size_t)gy) * 256 + (size_t)gx);
#if USE_ASYNC
    __builtin_amdgcn_global_load_async_to_lds_b32(
        (as1_i32)(uintptr_t)src,
        (as3_i32)(uint32_t)(uintptr_t)(&tile[i]),
        0, 0);
#else
    tile[i] = *src;
#endif
  }
#if USE_ASYNC
#if __has_builtin(__builtin_amdgcn_s_wait_asynccnt)
  __builtin_amdgcn_s_wait_asynccnt(0);
#else
  asm volatile("s_wait_asynccnt 0" ::: "memory");
#endif
#endif
  __syncthreads();

  // ---- Output 0: copy of x (write-once -> non-temporal) ----
  {
    float* oxb = out_x + (size_t)batch * (256ull * 256ull * 256ull);
    for (int i = tid; i < TILE * TILE * TILE; i += 256) {
      int lx = i & 15, ly = (i >> 4) & 15, lz = i >> 8;
      size_t g = (((size_t)(od + lz)) * 256 + (size_t)(oh + ly)) * 256 + (size_t)(ow + lx);
      float v = tile[((lz + HALO) * TLD + (ly + HALO)) * TLD + (lx + HALO)];
      __builtin_nontemporal_store(v, &oxb[g]);
    }
  }

  // ---- Output 1: 3x3x3 blur, stride 2 -> 8^3 outputs per tile ----
  {
    float* d1b = out_d1 + (size_t)batch * (128ull * 128ull * 128ull);
    for (int i = tid; i < 8 * 8 * 8; i += 256) {
      int lx = i & 7, ly = (i >> 3) & 7, lz = i >> 6;
      // window start in tile-local coords: 2*o - 1 (global) + HALO = 2*o + 1
      const float* base = &tile[((2 * lz + 1) * TLD + (2 * ly + 1)) * TLD + (2 * lx + 1)];
      float acc = 0.0f;
#pragma unroll
      for (int dz = 0; dz < 3; ++dz)
#pragma unroll
        for (int dy = 0; dy < 3; ++dy)
#pragma unroll
          for (int dx = 0; dx < 3; ++dx)
            acc = fmaf(k0s[(dz * 3 + dy) * 3 + dx],
                       base[(dz * TLD + dy) * TLD + dx], acc);
      size_t g = (((size_t)(bz * 8 + lz)) * 128 + (size_t)(bh * 8 + ly)) * 128
               + (size_t)(bw * 8 + lx);
      __builtin_nontemporal_store(acc, &d1b[g]);
    }
  }

  // ---- Output 2: 5x5x5 blur, stride 4 -> 4^3 outputs per tile ----
  {
    float* d2b = out_d2 + (size_t)batch * (64ull * 64ull * 64ull);
    for (int i = tid; i < 4 * 4 * 4; i += 256) {
      int lx = i & 3, ly = (i >> 2) & 3, lz = i >> 4;
      // window start in tile-local coords: 4*o - 2 (global) + HALO = 4*o
      const float* base = &tile[((4 * lz) * TLD + (4 * ly)) * TLD + (4 * lx)];
      float acc = 0.0f;
#pragma unroll
      for (int dz = 0; dz < 5; ++dz)
#pragma unroll
        for (int dy = 0; dy < 5; ++dy)
#pragma unroll
          for (int dx = 0; dx < 5; ++dx)
            acc = fmaf(k1s[(dz * 5 + dy) * 5 + dx],
                       base[(dz * TLD + dy) * TLD + dx], acc);
      size_t g = (((size_t)(bz * 4 + lz)) * 64 + (size_t)(bh * 4 + ly)) * 64
               + (size_t)(bw * 4 + lx);
      __builtin_nontemporal_store(acc, &d2b[g]);
    }
  }
}

extern "C" void kernel_launch(void* const* d_in, const int* in_sizes, int n_in,
                              void* d_out, int out_size, void* d_ws, size_t ws_size,
                              hipStream_t stream) {
  (void)in_sizes; (void)n_in; (void)out_size; (void)d_ws; (void)ws_size;

  const float* x  = (const float*)d_in[0];
  const float* k0 = (const float*)d_in[1];
  const float* k1 = (const float*)d_in[2];

  float* out    = (float*)d_out;
  float* out_x  = out;                                          // 2*256^3
  float* out_d1 = out_x + (size_t)2 * 256ull * 256ull * 256ull; // 2*128^3
  float* out_d2 = out_d1 + (size_t)2 * 128ull * 128ull * 128ull; // 2*64^3

  dim3 grid(16, 16, 32);   // W tiles, H tiles, D tiles * batch
  dim3 block(256);
  hipLaunchKernelGGL(pyr_fused_kernel, grid, block, 0, stream,
                     x, k0, k1, out_x, out_d1, out_d2);
}